// adqc_conv2d_82076825026863
// MI455X (gfx1250) — compile-verified
//
#include <hip/hip_runtime.h>
#include <hip/hip_bf16.h>

// CDNA5 / gfx1250, wave32.
typedef __attribute__((ext_vector_type(2))) float v2f;
typedef __attribute__((ext_vector_type(8))) float v8f;
typedef int v2i_ __attribute__((vector_size(8)));    // builtin param type (b64)
typedef int v4i_ __attribute__((vector_size(16)));   // builtin param type (b128)

#define IN_CH   64
#define OUT_CH  64
#define HW      128
#define NIMG    16
#define NGATES  45
#define STATE_N 65536   // 2^16 amplitudes

#define GAS __attribute__((address_space(1)))
#define LAS __attribute__((address_space(3)))

#if defined(__HIP_DEVICE_COMPILE__)
#if __has_builtin(__builtin_amdgcn_global_load_async_to_lds_b64) && \
    __has_builtin(__builtin_amdgcn_global_load_async_to_lds_b128)
#define HAVE_ASYNC 1
#endif
#endif

#if defined(HAVE_ASYNC)
#if __has_builtin(__builtin_amdgcn_s_wait_asynccnt)
#define WAITA(n) __builtin_amdgcn_s_wait_asynccnt(n)
#else
#define WAITA(n) asm volatile("s_wait_asynccnt %0" ::"i"(n) : "memory")
#endif
#else
#define WAITA(n)
#endif

// async (or fallback sync) copies: 8B and 16B per lane, global -> LDS.
// Pointer casts go through integers: AS1 is 64-bit, AS3 is the low 32 bits of
// the generic LDS address (ISA aperture rule).
__device__ __forceinline__ void cp_b64(float* ldst, const float* gsrc) {
#if defined(HAVE_ASYNC)
  __builtin_amdgcn_global_load_async_to_lds_b64(
      (GAS v2i_*)(unsigned long long)gsrc,
      (LAS v2i_*)(unsigned int)(unsigned long long)ldst, 0, 0);
#else
  ldst[0] = gsrc[0];
  ldst[1] = gsrc[1];
#endif
}
__device__ __forceinline__ void cp_b128(float* ldst, const float* gsrc) {
#if defined(HAVE_ASYNC)
  __builtin_amdgcn_global_load_async_to_lds_b128(
      (GAS v4i_*)(unsigned long long)gsrc,
      (LAS v4i_*)(unsigned int)(unsigned long long)ldst, 0, 0);
#else
  *(float4*)ldst = *(const float4*)gsrc;
#endif
}

// ---------------------------------------------------------------------------
// Kernel 1: synthesize conv weight from the 45 two-qubit gates.
// Whole 256KB state lives in dynamic LDS of a single workgroup (320KB WGP LDS).
// new[c] = sum_r G[r*4+c] * old[r] applied to bit-pair (b1+1, b1), b1 = 14-p0.
// ReLU after gates 14 and 29 (0-based). Output written as wT[k][oc],
// k = ic*16 + kh*4 + kw  (GEMM-ready: rows of 64 contiguous oc).
// ---------------------------------------------------------------------------
__global__ __launch_bounds__(256) void synth_weight_kernel(
    const float* __restrict__ gates, float* __restrict__ wT) {
  extern __shared__ float state[];  // 65536 floats = 256 KB
  const int tid = threadIdx.x;

  for (int i = tid; i < STATE_N; i += 256) state[i] = 0.0f;
  __syncthreads();
  if (tid == 0) state[0] = 1.0f;  // |0...0>
  __syncthreads();

  for (int g = 0; g < NGATES; ++g) {
    // brick-wall position: sweeps of 8,7,8,7,8,7 gates; p0 = (sweep%2) + 2*idx
    int idx = g, sweep = 0;
    while (true) {
      int cnt = (sweep & 1) ? 7 : 8;
      if (idx < cnt) break;
      idx -= cnt;
      ++sweep;
    }
    const int p0 = (sweep & 1) + 2 * idx;
    const int b1 = 14 - p0;            // lower bit of the pair
    const int lowMask = (1 << b1) - 1;

    float G[16];
#pragma unroll
    for (int e = 0; e < 16; ++e) G[e] = gates[g * 16 + e];

    for (int grp = tid; grp < STATE_N / 4; grp += 256) {
      const int low  = grp & lowMask;
      const int high = grp >> b1;
      const int base = (high << (b1 + 2)) | low;
      const float o0 = state[base];
      const float o1 = state[base | (1 << b1)];
      const float o2 = state[base | (2 << b1)];
      const float o3 = state[base | (3 << b1)];
      state[base]             = G[0] * o0 + G[4] * o1 + G[8]  * o2 + G[12] * o3;
      state[base | (1 << b1)] = G[1] * o0 + G[5] * o1 + G[9]  * o2 + G[13] * o3;
      state[base | (2 << b1)] = G[2] * o0 + G[6] * o1 + G[10] * o2 + G[14] * o3;
      state[base | (3 << b1)] = G[3] * o0 + G[7] * o1 + G[11] * o2 + G[15] * o3;
    }
    __syncthreads();

    if (g == 14 || g == 29) {  // ReLU after first two brick-wall layers
      for (int i = tid; i < STATE_N; i += 256)
        state[i] = fmaxf(state[i], 0.0f);
      __syncthreads();
    }
  }

  // state index i -> oc = i>>10, ic = (i>>4)&63, (kh,kw) = i&15
  for (int i = tid; i < STATE_N; i += 256) {
    const int oc = i >> 10;
    const int ic = (i >> 4) & 63;
    const int kk = i & 15;
    wT[(ic * 16 + kk) * OUT_CH + oc] = state[i];
  }
}

// ---------------------------------------------------------------------------
// Kernel 2: shifted conv as implicit GEMM with V_WMMA_F32_16X16X4_F32 and
// double-buffered async global->LDS staging (ASYNCcnt pipelined: tile ic+1
// copies overlap tile ic's 16 WMMAs; s_wait_asynccnt<=2 releases tile ic).
//
// 1 WG = one (image n, row y): 64 oc x 128 px. 8 waves = 4 M-tiles x 2 halves,
// each wave: 4 accumulators (16x16 f32). K loop: 64 ic x 4 kh, 4 kw per WMMA.
// Per ic we stage 4 input rows (cols 0..127 -> LDS cols 2..129; border cols
// pre-zeroed; out-of-range rows clamped at load and zeroed by a uniform
// select at use, so every wave issues exactly 2 async instructions per tile)
// plus the ic's 16x64 weight block.
// ---------------------------------------------------------------------------
__global__ __launch_bounds__(256) void conv_wmma_kernel(
    const float* __restrict__ x, const float* __restrict__ wT,
    float* __restrict__ out) {
  __shared__ __align__(16) float sIn[2][4 * 132];   // 2 x 2112 B
  __shared__ __align__(16) float sWt[2][16 * 64];   // 2 x 4096 B

  const int wg  = blockIdx.x;      // 0 .. 16*128-1
  const int n   = wg >> 7;
  const int y   = wg & 127;
  const int tid = threadIdx.x;
  const int lane = tid & 31;
  const int nidx = lane & 15;
  const int hi   = lane >> 4;      // 0 = K0/K1 half-wave, 1 = K2/K3
  const int ocBase = ((tid >> 5) & 3) * 16;
  const int pxBase = (tid >> 7) * 64;

  const float* xn = x + (size_t)n * IN_CH * HW * HW;

  // border columns (xs = -2,-1,128,129) are always conv/shift padding -> 0
  if (tid < 32) {
    const int b  = tid >> 4;
    const int rr = (tid >> 2) & 3;
    const int cc = tid & 3;
    sIn[b][rr * 132 + ((cc < 2) ? cc : cc + 128)] = 0.0f;
  }

  v8f acc[4];
#pragma unroll
  for (int s = 0; s < 4; ++s) acc[s] = (v8f){};

  // stage one tile: 4 rows x 128 cols input (b64/lane) + 16x64 weights (b128/lane)
  const int stR = tid >> 6;         // input row 0..3
  const int stC = (tid & 63) * 2;   // input col chunk (2 floats)
#define STAGE_TILE(ICV, BUF)                                                  \
  do {                                                                        \
    const int ic_ = (ICV);                                                    \
    const int sy_ = ((ic_ >> 4) < 2) ? 1 : 0;                                 \
    int ys_ = y + stR - 1 - sy_;                                              \
    if (ys_ < 0) ys_ = 0;                                                     \
    if (ys_ > HW - 1) ys_ = HW - 1;                                           \
    cp_b64(&sIn[BUF][stR * 132 + 2 + stC], xn + (ic_ * HW + ys_) * HW + stC); \
    cp_b128(&sWt[BUF][tid * 4], wT + ic_ * 16 * OUT_CH + tid * 4);            \
  } while (0)

  STAGE_TILE(0, 0);

  for (int ic = 0; ic < IN_CH; ++ic) {
    const int buf = ic & 1;
    __syncthreads();                 // everyone done reading buf^1 (tile ic-1)
    if (ic + 1 < IN_CH) {
      STAGE_TILE(ic + 1, buf ^ 1);   // overlap next tile's copies with compute
      WAITA(2);                      // in-order: tile ic's 2 asyncs complete
    } else {
      WAITA(0);
    }
    __syncthreads();                 // tile ic visible to all waves

    const int q  = ic >> 4;
    const int sy = (q < 2) ? 1 : 0;          // quarters 0,1 shift down
    const int sx = ((q & 1) == 0) ? 1 : 0;   // quarters 0,2 shift right
    const float* wtL = sWt[buf];
    const int coff = hi * 2 + 1 - sx;        // kw + 1 - sx (kw = hi*2, +1 for .y)

#pragma unroll
    for (int kh = 0; kh < 4; ++kh) {
      // A: 16x4 weight tile, K = ic*16 + kh*4 + {0..3}; lanes<16: K0/K1,
      // lanes>=16: K2/K3 (ISA 7.12.2 32-bit A layout)
      const int kb = kh * 4 + hi * 2;
      v2f A;
      A.x = wtL[(kb + 0) * OUT_CH + ocBase + nidx];
      A.y = wtL[(kb + 1) * OUT_CH + ocBase + nidx];

      // uniform row-validity select: zero out conv/shift padding rows
      const bool rv = ((unsigned)(y + kh - 1 - sy) < (unsigned)HW);
      const float* row = &sIn[buf][kh * 132];
#pragma unroll
      for (int s = 0; s < 4; ++s) {
        const int cb = pxBase + s * 16 + nidx + coff;
        v2f B;
        B.x = rv ? row[cb]     : 0.0f;
        B.y = rv ? row[cb + 1] : 0.0f;
        acc[s] = __builtin_amdgcn_wmma_f32_16x16x4_f32(
            false, A, false, B, (short)0, acc[s], false, false);
      }
    }
  }
#undef STAGE_TILE

  // C/D layout: VGPR j -> M = j + 8*hi, N = lane&15
  float* outn = out + (size_t)n * OUT_CH * HW * HW;
#pragma unroll
  for (int s = 0; s < 4; ++s) {
    const int px = pxBase + s * 16 + nidx;
#pragma unroll
    for (int j = 0; j < 8; ++j) {
      const int oc = ocBase + j + hi * 8;
      outn[(oc * HW + y) * HW + px] = acc[s][j];
    }
  }
}

// ---------------------------------------------------------------------------
extern "C" void kernel_launch(void* const* d_in, const int* in_sizes, int n_in,
                              void* d_out, int out_size, void* d_ws, size_t ws_size,
                              hipStream_t stream) {
  const float* x     = (const float*)d_in[0];  // [16,64,128,128] f32
  const float* gates = (const float*)d_in[1];  // [45,4,4] f32
  float* out = (float*)d_out;                  // [16,64,128,128] f32
  float* wT  = (float*)d_ws;                   // 1024*64 f32 = 256 KB scratch

  // 1 WG, 256 KB dynamic LDS: full state vector resident on one WGP
  synth_weight_kernel<<<1, 256, STATE_N * sizeof(float), stream>>>(gates, wT);

  // 2048 WGs (16 images x 128 rows), 8 wave32 per WG
  conv_wmma_kernel<<<NIMG * HW, 256, 0, stream>>>(x, wT, out);
}